// AGDNConv_14173392077052
// MI455X (gfx1250) — compile-verified
//
#include <hip/hip_runtime.h>

// ---------------------------------------------------------------------------
// AGDN conv for MI455X (gfx1250, wave32).
//   N=50000 nodes, E=1.6M edges, F_IN=256, H=2 heads, D=64, K=3 hops.
// Dense projections use V_WMMA_F32_16X16X4_F32 (full fp32 matrix pipe).
// SpMM hops: wave-per-edge float4 gather + f32 atomic scatter (BW-bound).
// ---------------------------------------------------------------------------

#define N_NODES  50000
#define N_EDGES  1600000
#define F_IN     256
#define H        2
#define D        64
#define HD       128          // H*D
#define KHOPS    3
#define NEG_SLOPE 0.2f
#define EPSV      1e-9f

typedef float v2f __attribute__((ext_vector_type(2)));
typedef float v8f __attribute__((ext_vector_type(8)));

// ---- wave32 all-lanes sum reduction -------------------------------------
__device__ __forceinline__ float wred32(float v) {
#pragma unroll
  for (int m = 16; m > 0; m >>= 1) v += __shfl_xor(v, m, 32);
  return v;
}

// ---- order-preserving float<->uint keys for exact atomic segment-max -----
__device__ __forceinline__ unsigned enc_f(float f) {
  unsigned b = __float_as_uint(f);
  return (b & 0x80000000u) ? ~b : (b | 0x80000000u);
}
__device__ __forceinline__ float dec_f(unsigned k) {
  unsigned b = (k & 0x80000000u) ? (k ^ 0x80000000u) : ~k;
  return __uint_as_float(b);
}

// ===========================================================================
// 1) Dense projections via WMMA f32 16x16x4.
//    One wave = one 16x16 tile of [h_src | h_dst] (256 output columns).
//    A lane L (half=L/16, r=L&15): A[r][k0+2*half+{0,1}], B[k0+2*half+{0,1}][r]
//    D: VGPR v holds row (v + 8*half), col r.
// ===========================================================================
__global__ void gemm_wmma_f32(const float* __restrict__ X,
                              const float* __restrict__ Wsrc,
                              const float* __restrict__ Wdst,
                              const float* __restrict__ bdst,
                              float* __restrict__ h_src,
                              float* __restrict__ h_dst) {
  const int lane   = threadIdx.x;        // 32 threads = 1 wave
  const int half   = lane >> 4;
  const int r      = lane & 15;
  const int tile_m = blockIdx.x * 16;
  const int tile_n = blockIdx.y * 16;    // 0..255: [0,128)=W_src, [128,256)=W_dst
  const bool is_dst = (tile_n >= HD);
  const float* Wm  = is_dst ? Wdst : Wsrc;
  const int c      = (is_dst ? (tile_n - HD) : tile_n) + r;

  v8f acc = {0.f, 0.f, 0.f, 0.f, 0.f, 0.f, 0.f, 0.f};
  const float* arow = X + (size_t)(tile_m + r) * F_IN;

#pragma unroll 4
  for (int k0 = 0; k0 < F_IN; k0 += 4) {
    const int ka = k0 + 2 * half;
    v2f a;
    {
      const float2 av = *reinterpret_cast<const float2*>(arow + ka);
      a.x = av.x; a.y = av.y;
    }
    v2f b;
    b.x = Wm[(size_t)ka * HD + c];
    b.y = Wm[(size_t)(ka + 1) * HD + c];
    acc = __builtin_amdgcn_wmma_f32_16x16x4_f32(false, a, false, b,
                                                (short)0, acc, false, false);
  }

  const float bias = is_dst ? bdst[c] : 0.f;
  float* Out = is_dst ? h_dst : h_src;
#pragma unroll
  for (int v = 0; v < 8; ++v)
    Out[(size_t)(tile_m + v + 8 * half) * HD + c] = acc[v] + bias;
}

// ===========================================================================
// 2) Skinny attention projections a_src/a_dst = X @ W_attn_* ([256,2]).
//    One wave per node; lanes stride K, shfl-reduce 4 sums.
// ===========================================================================
__global__ void attn_proj(const float* __restrict__ X,
                          const float* __restrict__ Was,
                          const float* __restrict__ Wad,
                          float* __restrict__ a_src,
                          float* __restrict__ a_dst) {
  const int n    = blockIdx.x * (blockDim.x >> 5) + (threadIdx.x >> 5);
  const int lane = threadIdx.x & 31;
  if (n >= N_NODES) return;
  const float* x = X + (size_t)n * F_IN;
  float s0 = 0.f, s1 = 0.f, d0 = 0.f, d1 = 0.f;
#pragma unroll
  for (int k = lane; k < F_IN; k += 32) {
    const float xv = x[k];
    s0 += xv * Was[k * 2 + 0];
    s1 += xv * Was[k * 2 + 1];
    d0 += xv * Wad[k * 2 + 0];
    d1 += xv * Wad[k * 2 + 1];
  }
  s0 = wred32(s0); s1 = wred32(s1); d0 = wred32(d0); d1 = wred32(d1);
  if (lane == 0) {
    a_src[n * 2 + 0] = s0; a_src[n * 2 + 1] = s1;
    a_dst[n * 2 + 0] = d0; a_dst[n * 2 + 1] = d1;
  }
}

// ===========================================================================
// 3) Edge softmax: three passes over E edges (logit+max, exp-sum, alpha).
// ===========================================================================
__global__ void edge_logits_max(const int* __restrict__ src,
                                const int* __restrict__ dst,
                                const float* __restrict__ a_src,
                                const float* __restrict__ a_dst,
                                float* __restrict__ ebuf,
                                unsigned* __restrict__ dmax,
                                unsigned* __restrict__ smax) {
  const int e = blockIdx.x * blockDim.x + threadIdx.x;
  if (e >= N_EDGES) return;
  const int s = src[e], d = dst[e];
#pragma unroll
  for (int h = 0; h < H; ++h) {
    float v = a_src[s * 2 + h] + a_dst[d * 2 + h];
    v = v > 0.f ? v : NEG_SLOPE * v;             // leaky_relu
    ebuf[(size_t)e * H + h] = v;
    const unsigned key = enc_f(v);
    atomicMax(&dmax[d * 2 + h], key);
    atomicMax(&smax[s * 2 + h], key);
  }
}

__global__ void edge_expsum(const int* __restrict__ src,
                            const int* __restrict__ dst,
                            const float* __restrict__ ebuf,
                            const unsigned* __restrict__ dmax,
                            const unsigned* __restrict__ smax,
                            float* __restrict__ dsum,
                            float* __restrict__ ssum) {
  const int e = blockIdx.x * blockDim.x + threadIdx.x;
  if (e >= N_EDGES) return;
  const int s = src[e], d = dst[e];
#pragma unroll
  for (int h = 0; h < H; ++h) {
    const float v = ebuf[(size_t)e * H + h];
    atomicAdd(&dsum[d * 2 + h], expf(v - dec_f(dmax[d * 2 + h])));
    atomicAdd(&ssum[s * 2 + h], expf(v - dec_f(smax[s * 2 + h])));
  }
}

__global__ void edge_alpha(const int* __restrict__ src,
                           const int* __restrict__ dst,
                           float* __restrict__ ebuf,   // in: logits, out: alpha
                           const unsigned* __restrict__ dmax,
                           const unsigned* __restrict__ smax,
                           const float* __restrict__ dsum,
                           const float* __restrict__ ssum) {
  const int e = blockIdx.x * blockDim.x + threadIdx.x;
  if (e >= N_EDGES) return;
  const int s = src[e], d = dst[e];
#pragma unroll
  for (int h = 0; h < H; ++h) {
    const float v  = ebuf[(size_t)e * H + h];
    const float pd = fmaxf(expf(v - dec_f(dmax[d * 2 + h])) / dsum[d * 2 + h], EPSV);
    const float ps = fmaxf(expf(v - dec_f(smax[s * 2 + h])) / ssum[s * 2 + h], EPSV);
    ebuf[(size_t)e * H + h] = sqrtf(pd * ps);
  }
}

// ===========================================================================
// 4) SpMM hop: wave per edge. Lane l covers 4 floats (head = l>>4).
//    B128 coalesced gather of h[src], 4x f32 atomic scatter to h_next[dst].
// ===========================================================================
__global__ void spmm_edge(const int* __restrict__ src,
                          const int* __restrict__ dst,
                          const float* __restrict__ alpha,
                          const float* __restrict__ hin,
                          float* __restrict__ hout) {
  const int wid  = blockIdx.x * (blockDim.x >> 5) + (threadIdx.x >> 5);
  const int lane = threadIdx.x & 31;
  if (wid >= N_EDGES) return;
  const int s = src[wid], d = dst[wid];
  const int h = lane >> 4;                        // lanes 0-15: head0, 16-31: head1
  const float w = alpha[(size_t)wid * H + h];
  const float4 v = *reinterpret_cast<const float4*>(hin + (size_t)s * HD + lane * 4);
  float* o = hout + (size_t)d * HD + lane * 4;
  atomicAdd(o + 0, v.x * w);
  atomicAdd(o + 1, v.y * w);
  atomicAdd(o + 2, v.z * w);
  atomicAdd(o + 3, v.w * w);
}

// ===========================================================================
// 5) feat_trans: per-(node,head) norm over D=64 + scale/offset/pos-emb.
//    Optionally fuse hop_attn_l dot (for k=0 / h0, which is never stored).
// ===========================================================================
__global__ void feat_trans(const float* __restrict__ hin,
                           float* __restrict__ hout,          // may be null
                           const float* __restrict__ scale,
                           const float* __restrict__ offset,
                           const float* __restrict__ pemb,
                           const int kidx,
                           const float* __restrict__ attn_l,  // may be null
                           float* __restrict__ a_l) {
  const int wid  = blockIdx.x * (blockDim.x >> 5) + (threadIdx.x >> 5);
  const int lane = threadIdx.x & 31;
  if (wid >= N_NODES * H) return;
  const int n = wid >> 1, h = wid & 1;
  const size_t base = (size_t)n * HD + h * D;
  const float x0 = hin[base + lane];
  const float x1 = hin[base + lane + 32];
  const float mean = wred32(x0 + x1) * (1.f / 64.f);
  const float m2   = wred32(x0 * x0 + x1 * x1) * (1.f / 64.f);
  const float rs   = rsqrtf(m2 - mean * mean + EPSV);
  const int pb = (kidx * H + h) * D;
  const float y0 = (x0 - mean) * scale[pb + lane] * rs + offset[pb + lane] + pemb[pb + lane];
  const float y1 = (x1 - mean) * scale[pb + lane + 32] * rs + offset[pb + lane + 32] + pemb[pb + lane + 32];
  if (hout) { hout[base + lane] = y0; hout[base + lane + 32] = y1; }
  if (attn_l) {
    const float al = wred32(y0 * attn_l[h * D + lane] + y1 * attn_l[h * D + lane + 32]);
    if (lane == 0) a_l[wid] = al;
  }
}

// ===========================================================================
// 6) Hop attention (softmax over K=3) + residual. Wave per (node,head).
// ===========================================================================
__global__ void hop_attn_final(const float* __restrict__ hs0,
                               const float* __restrict__ hs1,
                               const float* __restrict__ hs2,
                               const float* __restrict__ a_l,
                               const float* __restrict__ attn_r,
                               const float* __restrict__ h_dst,
                               float* __restrict__ out) {
  const int wid  = blockIdx.x * (blockDim.x >> 5) + (threadIdx.x >> 5);
  const int lane = threadIdx.x & 31;
  if (wid >= N_NODES * H) return;
  const int n = wid >> 1, h = wid & 1;
  const size_t base = (size_t)n * HD + h * D;
  const float v00 = hs0[base + lane], v01 = hs0[base + lane + 32];
  const float v10 = hs1[base + lane], v11 = hs1[base + lane + 32];
  const float v20 = hs2[base + lane], v21 = hs2[base + lane + 32];
  const float r0 = attn_r[h * D + lane], r1 = attn_r[h * D + lane + 32];
  const float al = a_l[wid];
  float l0 = wred32(v00 * r0 + v01 * r1) + al;
  float l1 = wred32(v10 * r0 + v11 * r1) + al;
  float l2 = wred32(v20 * r0 + v21 * r1) + al;
  l0 = l0 > 0.f ? l0 : NEG_SLOPE * l0;
  l1 = l1 > 0.f ? l1 : NEG_SLOPE * l1;
  l2 = l2 > 0.f ? l2 : NEG_SLOPE * l2;
  const float mx = fmaxf(l0, fmaxf(l1, l2));
  float w0 = expf(l0 - mx), w1 = expf(l1 - mx), w2 = expf(l2 - mx);
  const float inv = 1.f / (w0 + w1 + w2);
  w0 *= inv; w1 *= inv; w2 *= inv;
  out[base + lane]      = v00 * w0 + v10 * w1 + v20 * w2 + h_dst[base + lane];
  out[base + lane + 32] = v01 * w0 + v11 * w1 + v21 * w2 + h_dst[base + lane + 32];
}

// ===========================================================================
extern "C" void kernel_launch(void* const* d_in, const int* in_sizes, int n_in,
                              void* d_out, int out_size, void* d_ws, size_t ws_size,
                              hipStream_t stream) {
  (void)in_sizes; (void)n_in; (void)out_size; (void)ws_size;
  const float* feat   = (const float*)d_in[0];
  const int*   src    = (const int*)  d_in[1];
  const int*   dst    = (const int*)  d_in[2];
  const float* W_src  = (const float*)d_in[3];
  const float* W_dst  = (const float*)d_in[4];
  const float* b_dst  = (const float*)d_in[5];
  const float* W_as   = (const float*)d_in[6];
  const float* W_ad   = (const float*)d_in[7];
  const float* scale  = (const float*)d_in[8];
  const float* offset = (const float*)d_in[9];
  const float* pemb   = (const float*)d_in[10];
  const float* attn_l = (const float*)d_in[11];
  const float* attn_r = (const float*)d_in[12];
  float* out = (float*)d_out;

  // ---- workspace carve-up (~195 MB of fp32) ------------------------------
  const size_t NHD = (size_t)N_NODES * HD;
  const size_t NH  = (size_t)N_NODES * H;
  float* ws     = (float*)d_ws;
  float* h_src  = ws; ws += NHD;
  float* h_dstb = ws; ws += NHD;
  float* hbufA  = ws; ws += NHD;
  float* hbufB  = ws; ws += NHD;
  float* hs[KHOPS];
  for (int k = 0; k < KHOPS; ++k) { hs[k] = ws; ws += NHD; }
  float* ebuf   = ws; ws += (size_t)N_EDGES * H;   // logits, then alpha (in place)
  float* a_src  = ws; ws += NH;
  float* a_dst  = ws; ws += NH;
  float* a_l    = ws; ws += NH;
  float*    dsum = ws;            ws += NH;        // dsum|ssum|dmax|smax contiguous
  float*    ssum = ws;            ws += NH;
  unsigned* dmax = (unsigned*)ws; ws += NH;
  unsigned* smax = (unsigned*)ws; ws += NH;

  // zero segment accumulators (key 0 == -inf sentinel for encoded max)
  hipMemsetAsync(dsum, 0, 4 * NH * sizeof(float), stream);

  // ---- dense projections (WMMA) ------------------------------------------
  dim3 gg(N_NODES / 16, (2 * HD) / 16);            // (3125, 16), 1 wave/block
  gemm_wmma_f32<<<gg, 32, 0, stream>>>(feat, W_src, W_dst, b_dst, h_src, h_dstb);
  attn_proj<<<(N_NODES + 7) / 8, 256, 0, stream>>>(feat, W_as, W_ad, a_src, a_dst);

  // ---- edge softmax -------------------------------------------------------
  const int eb = (N_EDGES + 255) / 256;
  edge_logits_max<<<eb, 256, 0, stream>>>(src, dst, a_src, a_dst, ebuf, dmax, smax);
  edge_expsum   <<<eb, 256, 0, stream>>>(src, dst, ebuf, dmax, smax, dsum, ssum);
  edge_alpha    <<<eb, 256, 0, stream>>>(src, dst, ebuf, dmax, smax, dsum, ssum);

  // ---- h0 transform fused with hop_attn_l dot (h0 never materialized) ----
  const int nwb = (N_NODES * H + 7) / 8;
  feat_trans<<<nwb, 256, 0, stream>>>(h_src, nullptr, scale, offset, pemb, 0, attn_l, a_l);

  // ---- K diffusion hops (ping-pong raw h; hs[k] gets transformed copy) ---
  const float* hin = h_src;
  float* pp[KHOPS] = {hbufA, hbufB, hbufA};
  const int ewb = (N_EDGES + 7) / 8;               // 8 waves / 256-thread block
  for (int k = 0; k < KHOPS; ++k) {
    float* hnext = pp[k];
    hipMemsetAsync(hnext, 0, NHD * sizeof(float), stream);
    spmm_edge<<<ewb, 256, 0, stream>>>(src, dst, ebuf, hin, hnext);
    feat_trans<<<nwb, 256, 0, stream>>>(hnext, hs[k], scale, offset, pemb, k + 1,
                                        nullptr, nullptr);
    hin = hnext;
  }

  // ---- hop attention + residual ------------------------------------------
  hop_attn_final<<<nwb, 256, 0, stream>>>(hs[0], hs[1], hs[2], a_l, attn_r,
                                          h_dstb, out);
}